// ConvCaps_12610023981569
// MI455X (gfx1250) — compile-verified
//
#include <hip/hip_runtime.h>
#include <math.h>

typedef __attribute__((ext_vector_type(2))) float v2f;
typedef __attribute__((ext_vector_type(8))) float v8f;

#define NCAP 144
#define CCAP 32
#define PS 16
#define RSTRIDE 33
#define CEPS 1e-6f
#define CLAMBDA 0.01f
#define CLN_2PI 1.8378770664093453f

__global__ __launch_bounds__(256)
void convcaps_em_kernel(const float* __restrict__ x,
                        const float* __restrict__ a,
                        const float* __restrict__ W,
                        const float* __restrict__ beta_u,
                        const float* __restrict__ beta_a,
                        const int* __restrict__ iters_ptr,
                        float* __restrict__ out)
{
    // v: 144*32*16 fp32 = 288 KB (fits in 320 KB WGP LDS)
    __shared__ float sV[NCAP * CCAP * PS];
    // aliased buffer: im2col patch (2304 floats) during votes, then r[n][c] stride-33
    __shared__ float sPR[NCAP * RSTRIDE];
    __shared__ float sA[NCAP];
    __shared__ float sMu[CCAP * PS];
    __shared__ float sSig[CCAP * PS];
    __shared__ float sRsum[CCAP];
    __shared__ float sInv[CCAP];
    __shared__ float sAout[CCAP];

    const int tid  = threadIdx.x;
    const int lane = tid & 31;
    const int wave = tid >> 5;

    const int blk = blockIdx.x;
    const int bi  = blk / 196;
    const int pos = blk - bi * 196;
    const int oh  = pos / 14;
    const int ow  = pos - oh * 14;

    // ---- gather 3x3 patch of (16 caps x 16 pose) and activations into LDS ----
    // n = (kh*3 + kw)*16 + bc  (matches reference patching order)
    for (int e = tid; e < NCAP * PS; e += 256) {
        int n  = e >> 4, s = e & 15;
        int kh = n / 48;
        int kw = (n >> 4) % 3;
        int bc = n & 15;
        sPR[e] = x[(((bi * 16 + oh + kh) * 16 + (ow + kw)) * 16 + bc) * 16 + s];
    }
    for (int e = tid; e < NCAP; e += 256) {
        int kh = e / 48, kw = (e >> 4) % 3, bc = e & 15;
        sA[e] = a[((bi * 16 + oh + kh) * 16 + (ow + kw)) * 16 + bc];
    }
    __syncthreads();

    // ---- votes via V_WMMA_F32_16X16X4_F32 ----
    // Per n: A(16x4) rows = 16 (c,k) combos of w[n]^T, K=j.  B(4x16) cols = i (4 used).
    // D[m=(c,k), i] = sum_j w[n,c,j,k] * p[n,i,j] = v[n,c,i,k]
    {
        const int m       = lane & 15;          // A row / D col index base
        const int jb      = (lane >> 4) * 2;    // K split across lane halves
        const int col     = lane & 15;          // D column (= i)
        const int halfsel = lane >> 4;          // D row offset selector

        for (int n = wave * 18; n < wave * 18 + 18; ++n) {
            v2f bv;
            bv.x = (col < 4) ? sPR[n * 16 + col * 4 + jb]     : 0.0f;
            bv.y = (col < 4) ? sPR[n * 16 + col * 4 + jb + 1] : 0.0f;
            const float* Wn = W + n * (CCAP * PS);
#pragma unroll
            for (int t = 0; t < 8; ++t) {
                int g = t * 16 + m;
                int c = g >> 2, k = g & 3;
                v2f av;
                av.x = Wn[c * 16 + jb * 4 + k];
                av.y = Wn[c * 16 + jb * 4 + 4 + k];
                v8f acc = {0.f, 0.f, 0.f, 0.f, 0.f, 0.f, 0.f, 0.f};
                acc = __builtin_amdgcn_wmma_f32_16x16x4_f32(
                    false, av, false, bv, (short)0, acc, false, false);
                if (col < 4) {
#pragma unroll
                    for (int r = 0; r < 8; ++r) {
                        int m2 = r + 8 * halfsel;
                        int g2 = t * 16 + m2;
                        int c2 = g2 >> 2, k2 = g2 & 3;
                        sV[(n * CCAP + c2) * PS + col * 4 + k2] = acc[r];
                    }
                }
            }
        }
    }
    __syncthreads();

    // ---- EM routing (all LDS-resident) ----
    const int iters = iters_ptr[0];

    // init r[n,c] = a_in[n]/C   (overwrites the patch buffer)
    for (int e = tid; e < NCAP * CCAP; e += 256) {
        int n = e >> 5, c = e & 31;
        sPR[n * RSTRIDE + c] = sA[n] * (1.0f / 32.0f);
    }
    __syncthreads();

    for (int it = 0; it < iters; ++it) {
        // r_sum over n, per c
        if (tid < CCAP) {
            float ssum = 0.f;
            for (int n = 0; n < NCAP; ++n) ssum += sPR[n * RSTRIDE + tid];
            sRsum[tid] = ssum;
            sInv[tid]  = 1.0f / (ssum + CEPS);
        }
        __syncthreads();

        // mu[c,s] = sum_n coeff * v
        for (int idx = tid; idx < CCAP * PS; idx += 256) {
            int c = idx >> 4, s = idx & 15;
            float inv = sInv[c];
            float mv  = 0.f;
            for (int n = 0; n < NCAP; ++n)
                mv += sPR[n * RSTRIDE + c] * inv * sV[(n * CCAP + c) * PS + s];
            sMu[idx] = mv;
        }
        __syncthreads();

        // sigma_sq[c,s] = sum_n coeff*(v-mu)^2 + eps
        for (int idx = tid; idx < CCAP * PS; idx += 256) {
            int c = idx >> 4, s = idx & 15;
            float inv = sInv[c];
            float mu  = sMu[idx];
            float vv  = 0.f;
            for (int n = 0; n < NCAP; ++n) {
                float d = sV[(n * CCAP + c) * PS + s] - mu;
                vv += sPR[n * RSTRIDE + c] * inv * d * d;
            }
            sSig[idx] = vv + CEPS;
        }
        __syncthreads();

        // cost -> logit -> a_out
        if (tid < CCAP) {
            int   c    = tid;
            float bu   = beta_u[c];
            float cost = 0.f;
            for (int s = 0; s < PS; ++s)
                cost += bu + 0.5f * logf(sSig[c * PS + s]);
            cost *= sRsum[c];
            float logit = CLAMBDA * (beta_a[c] - cost);
            sAout[c] = 1.0f / (1.0f + expf(-logit));
        }
        __syncthreads();

        if (it < iters - 1) {
            // ln_ap[n,c] (overwrite r buffer; nothing reads r here)
            for (int e = tid; e < NCAP * CCAP; e += 256) {
                int n = e >> 5, c = e & 31;
                float lp = 0.f;
                for (int s = 0; s < PS; ++s) {
                    float sg = sSig[c * PS + s];
                    float d  = sV[(n * CCAP + c) * PS + s] - sMu[c * PS + s];
                    lp += -(d * d) / (2.0f * sg) - 0.5f * logf(sg);
                }
                lp += -8.0f * CLN_2PI + logf(sAout[c] + CEPS);
                sPR[n * RSTRIDE + c] = lp;
            }
            __syncthreads();
            // softmax over c, then * a_in[n]
            if (tid < NCAP) {
                int n = tid;
                float mx = -3.0e38f;
                for (int c = 0; c < CCAP; ++c)
                    mx = fmaxf(mx, sPR[n * RSTRIDE + c]);
                float se = 0.f;
                for (int c = 0; c < CCAP; ++c)
                    se += expf(sPR[n * RSTRIDE + c] - mx);
                float scale = sA[n] / se;
                for (int c = 0; c < CCAP; ++c)
                    sPR[n * RSTRIDE + c] = expf(sPR[n * RSTRIDE + c] - mx) * scale;
            }
        }
        __syncthreads();
    }

    // ---- outputs: p_out (b,14,14,32,16) then a_out (b,14,14,32,1), concatenated ----
    const int pbase = blk * (CCAP * PS);
    for (int idx = tid; idx < CCAP * PS; idx += 256)
        out[pbase + idx] = sMu[idx];
    float* abase = out + (size_t)gridDim.x * (CCAP * PS);
    for (int c = tid; c < CCAP; c += 256)
        abase[blk * CCAP + c] = sAout[c];
}

extern "C" void kernel_launch(void* const* d_in, const int* in_sizes, int n_in,
                              void* d_out, int out_size, void* d_ws, size_t ws_size,
                              hipStream_t stream) {
    const float* x  = (const float*)d_in[0];
    const float* a  = (const float*)d_in[1];
    const float* W  = (const float*)d_in[2];
    const float* bu = (const float*)d_in[3];
    const float* ba = (const float*)d_in[4];
    const int* iters = (const int*)d_in[5];
    float* out = (float*)d_out;

    int b = in_sizes[0] / (16 * 16 * 16 * 16);   // batch from x element count
    dim3 grid(b * 14 * 14);                      // one workgroup per output position
    convcaps_em_kernel<<<grid, 256, 0, stream>>>(x, a, W, bu, ba, iters, out);
}